// Seq2Seq_56092272886280
// MI455X (gfx1250) — compile-verified
//
#include <hip/hip_runtime.h>
#include <math.h>

#define V_  10000
#define E_  128
#define H1_ 128
#define H2_ 64
#define D_  64
#define T_  512
#define B_  256
#define MAXLEN_ 30
#define SOS_ 1
#define EOS_ 2

typedef float v2f __attribute__((ext_vector_type(2)));
typedef float v8f __attribute__((ext_vector_type(8)));

__device__ __forceinline__ float sigmoidf_(float x) {
    return 1.0f / (1.0f + __expf(-x));
}

// Accumulate a 16x16 f32 tile: acc += A(16xK) * W(16xK)^T using
// V_WMMA_F32_16X16X4_F32. A: row-major [m][k] with leading dim ldA (LDS).
// W: row-major weight rows [n][k] with leading dim ldW, pre-offset to row n0
// (each lane adds lo*ldW). Lane mapping per CDNA5 ISA 32-bit A/B layouts:
//   A: lane lo holds M=lo, K = k + 2*half + {0,1}
//   B: lane lo holds N=lo, K = k + 2*half + {0,1}
__device__ __forceinline__ v8f mm_acc(v8f acc,
                                      const float* __restrict__ A, int ldA,
                                      const float* __restrict__ W, int ldW,
                                      int K, int lo, int half) {
    const float* a_ptr = A + lo * ldA + 2 * half;
    const float* w_ptr = W + (size_t)lo * ldW + 2 * half;
    #pragma unroll 4
    for (int k = 0; k < K; k += 4) {
        v2f a = *(const v2f*)(a_ptr + k);
        v2f b = *(const v2f*)(w_ptr + k);
        acc = __builtin_amdgcn_wmma_f32_16x16x4_f32(
            false, a, false, b, (short)0, acc, false, false);
    }
    return acc;
}

// ---------------------------------------------------------------------------
// Init: zero recurrent state, tok = SOS (argmax of pred0).
// ---------------------------------------------------------------------------
__global__ void init_kernel(float* __restrict__ Z, float* __restrict__ h1,
                            float* __restrict__ c1, float* __restrict__ c2,
                            int* __restrict__ tok) {
    int i = blockIdx.x * blockDim.x + threadIdx.x;
    if (i < B_ * 128) { Z[i] = 0.0f; h1[i] = 0.0f; c1[i] = 0.0f; }
    if (i < B_ * 64)  c2[i] = 0.0f;
    if (i < B_)       tok[i] = SOS_;
}

// ---------------------------------------------------------------------------
// RNN step: embedding lookup (EOS row zeroed), LSTM1 (H1=128, in=192),
// LSTM2 (H2=64, in=128), query = h2 @ wq^T + bq.  One WG per 16 batch rows.
// All GEMMs via f32 WMMA, elementwise gates in LDS.
// ---------------------------------------------------------------------------
__global__ __launch_bounds__(256) void rnn_kernel(
    const int* __restrict__ tok, const float* __restrict__ emb,
    const float* __restrict__ w_ih1, const float* __restrict__ w_hh1,
    const float* __restrict__ b_ih1, const float* __restrict__ b_hh1,
    const float* __restrict__ w_ih2, const float* __restrict__ w_hh2,
    const float* __restrict__ b_ih2, const float* __restrict__ b_hh2,
    const float* __restrict__ wq, const float* __restrict__ bq,
    float* __restrict__ Z,   // (B,128): [h2 | context]
    float* __restrict__ h1, float* __restrict__ c1, float* __restrict__ c2,
    float* __restrict__ query) {
    __shared__ float s_inp[16 * 192];   // [emb | context]
    __shared__ float s_h1[16 * 128];
    __shared__ float s_h2[16 * 64];
    __shared__ float s_g[16 * 512];     // gates scratch (reused for LSTM2)

    const int tid  = threadIdx.x;
    const int m0   = blockIdx.x * 16;
    const int wave = tid >> 5;
    const int lane = tid & 31;
    const int lo   = lane & 15;
    const int half = lane >> 4;

    // ---- phase 0: gather inputs into LDS ----
    for (int i = tid; i < 16 * 192; i += 256) {
        int m = i / 192, k = i % 192;
        float v;
        if (k < E_) {
            int t = tok[m0 + m];
            v = (t == EOS_) ? 0.0f : emb[(size_t)t * E_ + k];  // padding_idx=EOS
        } else {
            v = Z[(m0 + m) * 128 + 64 + (k - E_)];             // old context
        }
        s_inp[m * 192 + k] = v;
    }
    for (int i = tid; i < 16 * 128; i += 256) s_h1[i] = h1[m0 * 128 + i];
    for (int i = tid; i < 16 * 64; i += 256) {
        int m = i >> 6, j = i & 63;
        s_h2[i] = Z[(m0 + m) * 128 + j];                       // old h2
    }
    __syncthreads();

    // ---- LSTM1 gates: (16x192)(192x512) + (16x128)(128x512), 32 N-tiles ----
    for (int nt = wave; nt < 32; nt += 8) {
        int n0 = nt * 16;
        v8f acc = {};
        acc = mm_acc(acc, s_inp, 192, w_ih1 + (size_t)n0 * 192, 192, 192, lo, half);
        acc = mm_acc(acc, s_h1, 128, w_hh1 + (size_t)n0 * 128, 128, 128, lo, half);
        float bias = b_ih1[n0 + lo] + b_hh1[n0 + lo];
        #pragma unroll
        for (int r = 0; r < 8; ++r)
            s_g[(r + 8 * half) * 512 + n0 + lo] = acc[r] + bias;
    }
    __syncthreads();

    // ---- LSTM1 elementwise (gate order i,f,g,o) ----
    for (int i = tid; i < 16 * 128; i += 256) {
        int m = i >> 7, j = i & 127;
        float gi = s_g[m * 512 + j];
        float gf = s_g[m * 512 + 128 + j];
        float gg = s_g[m * 512 + 256 + j];
        float go = s_g[m * 512 + 384 + j];
        float c  = c1[(m0 + m) * 128 + j];
        float cn = sigmoidf_(gf) * c + sigmoidf_(gi) * tanhf(gg);
        float hn = sigmoidf_(go) * tanhf(cn);
        c1[(m0 + m) * 128 + j] = cn;
        h1[(m0 + m) * 128 + j] = hn;
        s_h1[m * 128 + j] = hn;        // new h1 feeds LSTM2
    }
    __syncthreads();

    // ---- LSTM2 gates: (16x128)(128x256) + (16x64)(64x256), 16 N-tiles ----
    for (int nt = wave; nt < 16; nt += 8) {
        int n0 = nt * 16;
        v8f acc = {};
        acc = mm_acc(acc, s_h1, 128, w_ih2 + (size_t)n0 * 128, 128, 128, lo, half);
        acc = mm_acc(acc, s_h2, 64, w_hh2 + (size_t)n0 * 64, 64, 64, lo, half);
        float bias = b_ih2[n0 + lo] + b_hh2[n0 + lo];
        #pragma unroll
        for (int r = 0; r < 8; ++r)
            s_g[(r + 8 * half) * 256 + n0 + lo] = acc[r] + bias;
    }
    __syncthreads();

    // ---- LSTM2 elementwise ----
    for (int i = tid; i < 16 * 64; i += 256) {
        int m = i >> 6, j = i & 63;
        float gi = s_g[m * 256 + j];
        float gf = s_g[m * 256 + 64 + j];
        float gg = s_g[m * 256 + 128 + j];
        float go = s_g[m * 256 + 192 + j];
        float c  = c2[(m0 + m) * 64 + j];
        float cn = sigmoidf_(gf) * c + sigmoidf_(gi) * tanhf(gg);
        float hn = sigmoidf_(go) * tanhf(cn);
        c2[(m0 + m) * 64 + j] = cn;
        s_h2[m * 64 + j] = hn;
        Z[(m0 + m) * 128 + j] = hn;    // new h2 into Z[:, :64]
    }
    __syncthreads();

    // ---- query = h2_new @ wq^T + bq : 4 N-tiles, waves 0..3 ----
    if (wave < 4) {
        int n0 = wave * 16;
        v8f acc = {};
        acc = mm_acc(acc, s_h2, 64, wq + (size_t)n0 * 64, 64, 64, lo, half);
        float bias = bq[n0 + lo];
        #pragma unroll
        for (int r = 0; r < 8; ++r)
            query[(m0 + r + 8 * half) * 64 + n0 + lo] = acc[r] + bias;
    }
}

// ---------------------------------------------------------------------------
// Attention: one WG per batch row. energy over T=512, masked renormalized
// softmax:  w_t = mask_t * exp(e_t - mx) / max(S_masked, 2e-30 * S_all),
// context[d] = sum_t w_t * enc_values[t,b,d].  Writes Z[b, 64:128].
// ---------------------------------------------------------------------------
__global__ __launch_bounds__(256) void attn_kernel(
    const float* __restrict__ enc_key, const float* __restrict__ enc_values,
    const int* __restrict__ mask, const float* __restrict__ query,
    float* __restrict__ Z) {
    __shared__ float s_q[64];
    __shared__ float s_e[T_];
    __shared__ float s_red[256];
    __shared__ float s_ctx[4][64];

    const int b = blockIdx.x;
    const int tid = threadIdx.x;

    if (tid < 64) s_q[tid] = query[b * 64 + tid];
    __syncthreads();

    // energy[t] = q . enc_key[t, b, :]
    for (int t = tid; t < T_; t += 256) {
        const float* ek = enc_key + ((size_t)t * B_ + b) * D_;
        float e = 0.0f;
        #pragma unroll
        for (int d = 0; d < D_; d += 4) {
            float4 k4 = *(const float4*)(ek + d);
            e += s_q[d] * k4.x + s_q[d + 1] * k4.y +
                 s_q[d + 2] * k4.z + s_q[d + 3] * k4.w;
        }
        s_e[t] = e;
    }
    __syncthreads();

    // max reduce
    float mx = -INFINITY;
    for (int t = tid; t < T_; t += 256) mx = fmaxf(mx, s_e[t]);
    s_red[tid] = mx;
    __syncthreads();
    for (int s = 128; s > 0; s >>= 1) {
        if (tid < s) s_red[tid] = fmaxf(s_red[tid], s_red[tid + s]);
        __syncthreads();
    }
    mx = s_red[0];
    __syncthreads();

    // exp + both sums (overwrite s_e with masked exp)
    float se_all = 0.0f, se_m = 0.0f;
    for (int t = tid; t < T_; t += 256) {
        float ex = __expf(s_e[t] - mx);
        float mk = (float)mask[(size_t)b * T_ + t];
        s_e[t] = ex * mk;
        se_all += ex;
        se_m += ex * mk;
    }
    s_red[tid] = se_all;
    __syncthreads();
    for (int s = 128; s > 0; s >>= 1) {
        if (tid < s) s_red[tid] += s_red[tid + s];
        __syncthreads();
    }
    se_all = s_red[0];
    __syncthreads();
    s_red[tid] = se_m;
    __syncthreads();
    for (int s = 128; s > 0; s >>= 1) {
        if (tid < s) s_red[tid] += s_red[tid + s];
        __syncthreads();
    }
    se_m = s_red[0];
    __syncthreads();

    float inv = 1.0f / fmaxf(se_m, 2e-30f * se_all);

    // context: thread -> (slice of T, one d); coalesced over d
    int d = tid & 63, sl = tid >> 6;
    float acc = 0.0f;
    for (int t = sl * 128; t < (sl + 1) * 128; ++t)
        acc += s_e[t] * enc_values[((size_t)t * B_ + b) * D_ + d];
    s_ctx[sl][d] = acc;
    __syncthreads();
    if (tid < 64) {
        float c = (s_ctx[0][tid] + s_ctx[1][tid] + s_ctx[2][tid] + s_ctx[3][tid]) * inv;
        Z[b * 128 + 64 + tid] = c;
    }
}

// ---------------------------------------------------------------------------
// Output projection: pred = [h2|ctx](256x128) @ wp^T(128x10000) + bp,
// written straight into d_out[b, step, :].  WMMA f32, 16x16 tiles.
// grid = (79 blocks of 8 wave-tiles -> 632 >= 625 N-tiles, clamped), 16 M-tiles.
// ---------------------------------------------------------------------------
__global__ __launch_bounds__(256) void pred_kernel(
    const float* __restrict__ Z, const float* __restrict__ wp,
    const float* __restrict__ bp, float* __restrict__ out, int step) {
    __shared__ float s_z[16 * 128];
    const int tid  = threadIdx.x;
    const int m0   = blockIdx.y * 16;
    const int wave = tid >> 5;
    const int lane = tid & 31;
    const int lo   = lane & 15;
    const int half = lane >> 4;

    for (int i = tid; i < (16 * 128) / 4; i += 256)
        ((float4*)s_z)[i] = ((const float4*)(Z + m0 * 128))[i];
    __syncthreads();

    int tile = blockIdx.x * 8 + wave;
    if (tile > 624) tile = 624;      // duplicate tile: identical stores, benign
    int n0 = tile * 16;

    v8f acc = {};
    acc = mm_acc(acc, s_z, 128, wp + (size_t)n0 * 128, 128, 128, lo, half);
    float bias = bp[n0 + lo];
    #pragma unroll
    for (int r = 0; r < 8; ++r) {
        int m = m0 + r + 8 * half;
        out[((size_t)m * MAXLEN_ + step) * V_ + n0 + lo] = acc[r] + bias;
    }
}

// ---------------------------------------------------------------------------
// Greedy argmax over V=10000 per row, first-occurrence tie-break (jnp.argmax).
// ---------------------------------------------------------------------------
__global__ __launch_bounds__(256) void argmax_kernel(
    const float* __restrict__ out, int step, int* __restrict__ tok) {
    __shared__ float s_v[256];
    __shared__ int s_i[256];
    const int b = blockIdx.x;
    const int tid = threadIdx.x;
    const float* p = out + ((size_t)b * MAXLEN_ + step) * V_;

    float best = -INFINITY;
    int bi = 0x7fffffff;
    for (int v = tid; v < V_; v += 256) {
        float x = p[v];
        if (x > best) { best = x; bi = v; }   // within-thread: first (lowest v)
    }
    s_v[tid] = best;
    s_i[tid] = bi;
    __syncthreads();
    for (int s = 128; s > 0; s >>= 1) {
        if (tid < s) {
            float xv = s_v[tid + s];
            int xi = s_i[tid + s];
            if (xv > s_v[tid] || (xv == s_v[tid] && xi < s_i[tid])) {
                s_v[tid] = xv;
                s_i[tid] = xi;
            }
        }
        __syncthreads();
    }
    if (tid == 0) tok[b] = s_i[0];
}

// ---------------------------------------------------------------------------
extern "C" void kernel_launch(void* const* d_in, const int* in_sizes, int n_in,
                              void* d_out, int out_size, void* d_ws, size_t ws_size,
                              hipStream_t stream) {
    const float* enc_key    = (const float*)d_in[0];
    const float* enc_values = (const float*)d_in[1];
    const int*   mask       = (const int*)d_in[2];
    const float* emb        = (const float*)d_in[3];
    const float* w_ih1      = (const float*)d_in[4];
    const float* w_hh1      = (const float*)d_in[5];
    const float* b_ih1      = (const float*)d_in[6];
    const float* b_hh1      = (const float*)d_in[7];
    const float* w_ih2      = (const float*)d_in[8];
    const float* w_hh2      = (const float*)d_in[9];
    const float* b_ih2      = (const float*)d_in[10];
    const float* b_hh2      = (const float*)d_in[11];
    const float* wq         = (const float*)d_in[12];
    const float* bq         = (const float*)d_in[13];
    const float* wp         = (const float*)d_in[14];
    const float* bp         = (const float*)d_in[15];
    float* out = (float*)d_out;

    // workspace layout (floats): Z | h1 | c1 | c2 | query | tok(int)
    float* ws    = (float*)d_ws;
    float* Z     = ws;                  // 256*128
    float* h1    = Z + 256 * 128;       // 256*128
    float* c1    = h1 + 256 * 128;      // 256*128
    float* c2    = c1 + 256 * 128;      // 256*64
    float* query = c2 + 256 * 64;       // 256*64
    int*   tok   = (int*)(query + 256 * 64);  // 256

    init_kernel<<<128, 256, 0, stream>>>(Z, h1, c1, c2, tok);
    for (int step = 0; step < MAXLEN_; ++step) {
        rnn_kernel<<<16, 256, 0, stream>>>(tok, emb, w_ih1, w_hh1, b_ih1, b_hh1,
                                           w_ih2, w_hh2, b_ih2, b_hh2, wq, bq,
                                           Z, h1, c1, c2, query);
        attn_kernel<<<B_, 256, 0, stream>>>(enc_key, enc_values, mask, query, Z);
        pred_kernel<<<dim3(79, 16), 256, 0, stream>>>(Z, wp, bp, out, step);
        if (step + 1 < MAXLEN_)
            argmax_kernel<<<B_, 256, 0, stream>>>(out, step, tok);
    }
}